// FinalLayer_multioutput_54881092108541
// MI455X (gfx1250) — compile-verified
//
#include <hip/hip_runtime.h>

#define NB     8
#define NG     512
#define NT     512
#define NBASIS 5
#define NCH    3
#define NKC    15          // nbasis * nchannel
#define TT     16          // targets per block
#define GP     16          // g-partitions per block
#define GPG    (NG / GP)   // 32 grid points per partition
#define EPSF   1e-6f
#define LOG2E  1.4426950408889634f

typedef __attribute__((ext_vector_type(2))) float v2f;
typedef __attribute__((ext_vector_type(8))) float v8f;

// Async global->LDS copy of one b32 per lane (CDNA5 GVS mode):
//   copies one dword from the global address saddr + voff to the LDS address in lds_off
// Tracked on ASYNCcnt; caller must s_wait_asynccnt before consuming.
__device__ __forceinline__ void async_b32_to_lds(unsigned lds_off, unsigned voff,
                                                 const void* saddr) {
    asm volatile("global_load_async_to_lds_b32 %0, %1, %2 offset:0"
                 :: "v"(lds_off), "v"(voff), "s"(saddr) : "memory");
}
__device__ __forceinline__ void wait_asynccnt0() {
    asm volatile("s_wait_asynccnt 0" ::: "memory");
}

__global__ __launch_bounds__(256)
void convcnp_final_kernel(const float* __restrict__ x_grid,
                          const float* __restrict__ h_grid,
                          const float* __restrict__ target_x,
                          const float* __restrict__ sigma,
                          const float* __restrict__ g_w,
                          const float* __restrict__ g_b,
                          float* __restrict__ out)
{
    __shared__ float xs[NG * NCH];        // 6 KB   x_grid[b] tile
    __shared__ float hs[NG * 16];         // 32 KB  h_grid[b], n padded 15->16
    __shared__ float m2s[NKC];            // -0.5*log2e / s^2 per (k,c)
    __shared__ float BmT[16 * 16];        // 1 KB   B transposed: BmT[c][n] = B[n][c]
    __shared__ float P[GP * TT * 17];     // 17 KB  per-partition partials (pad 17)
    __shared__ float H[TT * 16];          // 1 KB   reduced h_out tile (A of final GEMM)

    const int tid = threadIdx.x;
    const int b   = blockIdx.x / (NT / TT);
    const int t0  = (blockIdx.x % (NT / TT)) * TT;

    // --- per-(k,c) exponent constants ---
    if (tid < NKC) {
        float s = __expf(sigma[tid]) + EPSF;      // scales, (k,c) row-major
        m2s[tid] = -0.5f * LOG2E / (s * s);       // wt = exp2(m2 * d^2)
    }
    // --- B matrix of final projection, transposed for contiguous b64 reads ---
    {
        int c = tid >> 4, n = tid & 15;           // 256 entries
        float v = 0.f;
        if (n < NKC && (n % 3) == c) v = g_w[n / 3];
        BmT[tid] = v;
    }
    // --- async-stage x_grid[b], h_grid[b] into LDS (no VGPR round trip) ---
    const float* xg = x_grid + (size_t)b * NG * NCH;
    const float* hg = h_grid + (size_t)b * NG * NKC;
    const unsigned xs_base = (unsigned)(uintptr_t)&xs[0];
    const unsigned hs_base = (unsigned)(uintptr_t)&hs[0];
    for (int i = tid; i < NG * NCH; i += 256)
        async_b32_to_lds(xs_base + 4u * i, 4u * i, xg);
    for (int i = tid; i < NG * NKC; i += 256) {
        int g = i / NKC, n = i - g * NKC;
        async_b32_to_lds(hs_base + 4u * (g * 16 + n), 4u * i, hg);
    }
    wait_asynccnt0();
    __syncthreads();

    // --- main phase: thread = (t_local, g-partition) ---
    const int t_local = tid & (TT - 1);
    const int gp      = tid >> 4;

    float y0, y1, y2;
    {
        const float* ty = target_x + ((size_t)b * NT + t0 + t_local) * NCH;
        y0 = ty[0]; y1 = ty[1]; y2 = ty[2];
    }
    float m2[NKC];
#pragma unroll
    for (int n = 0; n < NKC; ++n) m2[n] = m2s[n];

    float acc[NKC];
#pragma unroll
    for (int n = 0; n < NKC; ++n) acc[n] = 0.f;

    const int gbeg = gp * GPG;
    for (int g = gbeg; g < gbeg + GPG; ++g) {
        float d0 = xs[g * 3 + 0] - y0;
        float d1 = xs[g * 3 + 1] - y1;
        float d2 = xs[g * 3 + 2] - y2;
        float q0 = d0 * d0, q1 = d1 * d1, q2 = d2 * d2;
#pragma unroll
        for (int n = 0; n < NKC; ++n) {
            float dq = (n % 3 == 0) ? q0 : ((n % 3 == 1) ? q1 : q2);
            float w  = __builtin_amdgcn_exp2f(m2[n] * dq);   // v_exp_f32
            acc[n]  += w * hs[g * 16 + n];
        }
    }

    // --- write partials (P[gp][t][*17] -> conflict-free) ---
#pragma unroll
    for (int n = 0; n < NKC; ++n)
        P[(gp * TT + t_local) * 17 + n] = acc[n];
    __syncthreads();

    // --- reduce over g-partitions: thread = (t_r, n_r) ---
    {
        int n_r = tid & 15, t_r = tid >> 4;
        float s = 0.f;
        if (n_r < NKC) {
#pragma unroll
            for (int gpp = 0; gpp < GP; ++gpp)
                s += P[(gpp * TT + t_r) * 17 + n_r];
        }
        H[t_r * 16 + n_r] = s;        // n=15 column is zero padding
    }
    __syncthreads();

    // --- final projection: out[t,c] = sum_n H[t,n]*B[n,c] + g_b ---
    // 4 chained V_WMMA_F32_16X16X4_F32 by wave 0 (EXEC all ones in that wave).
    if (tid < 32) {
        const int lane = tid;
        const int half = lane >> 4;   // lanes 0-15: K=+0/+1 ; lanes 16-31: K=+2/+3
        const int l15  = lane & 15;   // A: M row / B: N column

        v8f dacc = {};                // C = 0
#pragma unroll
        for (int q = 0; q < 4; ++q) {
            const int ka = 4 * q + 2 * half;
            v2f a, bb;
            a.x  = H[l15 * 16 + ka];          // ds_load_b64 (aligned pair)
            a.y  = H[l15 * 16 + ka + 1];
            bb.x = BmT[l15 * 16 + ka];        // ds_load_b64 (aligned pair)
            bb.y = BmT[l15 * 16 + ka + 1];
            dacc = __builtin_amdgcn_wmma_f32_16x16x4_f32(
                false, a, false, bb, (short)0, dacc, false, false);
        }

        const float bias = g_b[0];
        // D layout: VGPR r, lanes 0-15 -> (M=r, N=lane); lanes 16-31 -> (M=8+r, N=lane-16)
        if (l15 < NCH) {
#pragma unroll
            for (int r = 0; r < 8; ++r) {
                int m = half * 8 + r;
                out[((size_t)b * NT + t0 + m) * NCH + l15] = dacc[r] + bias;
            }
        }
    }
}

extern "C" void kernel_launch(void* const* d_in, const int* in_sizes, int n_in,
                              void* d_out, int out_size, void* d_ws, size_t ws_size,
                              hipStream_t stream) {
    (void)in_sizes; (void)n_in; (void)out_size; (void)d_ws; (void)ws_size;
    const float* x_grid   = (const float*)d_in[0];
    const float* h_grid   = (const float*)d_in[1];
    const float* target_x = (const float*)d_in[2];
    const float* sigma    = (const float*)d_in[3];
    const float* g_w      = (const float*)d_in[4];
    const float* g_b      = (const float*)d_in[5];
    float* out = (float*)d_out;

    dim3 grid(NB * (NT / TT));   // 8 * 32 = 256 blocks
    dim3 block(256);             // 8 wave32 per block
    convcnp_final_kernel<<<grid, block, 0, stream>>>(
        x_grid, h_grid, target_x, sigma, g_w, g_b, out);
}